// DeepMOI_18863496364776
// MI455X (gfx1250) — compile-verified
//
#include <hip/hip_runtime.h>
#include <math.h>

typedef float v2f __attribute__((ext_vector_type(2)));
typedef float v8f __attribute__((ext_vector_type(8)));

__device__ __forceinline__ float sigmoidf_(float x) { return 1.0f / (1.0f + __expf(-x)); }

// ---------------------------------------------------------------- zero scratch
__global__ void zero_f32(float* p, int n) {
    int i = blockIdx.x * blockDim.x + threadIdx.x;
    for (; i < n; i += gridDim.x * blockDim.x) p[i] = 0.0f;
}

// ---------------------------------------------------------------- edge scatter: agg[dst] += x[src]
template <int DM>
__global__ void scatter_add_k(const float* __restrict__ x,
                              const int* __restrict__ src,
                              const int* __restrict__ dst,
                              float* __restrict__ agg, int E) {
    int e = blockIdx.x * blockDim.x + threadIdx.x;
    if (e >= E) return;
    int s = src[e];
    int d = dst[e];
    const float* xs = x + (size_t)s * DM;
    float* ad = agg + (size_t)d * DM;
#pragma unroll
    for (int j = 0; j < DM; ++j) atomicAdd(&ad[j], xs[j]);
}

// ---------------------------------------------------------------- GIN apply: y = relu(W*(x+agg)+b)
template <int DIN, int DOUT>
__global__ void gin_apply_k(const float* __restrict__ x,
                            const float* __restrict__ agg,
                            const float* __restrict__ W,   // [DOUT, DIN] row-major
                            const float* __restrict__ b,
                            float* __restrict__ y, int N) {
    int i = blockIdx.x * blockDim.x + threadIdx.x;
    if (i >= N) return;
    float v[DIN];
#pragma unroll
    for (int d = 0; d < DIN; ++d) v[d] = x[(size_t)i * DIN + d] + agg[(size_t)i * DIN + d];
#pragma unroll
    for (int j = 0; j < DOUT; ++j) {
        float acc = b[j];
#pragma unroll
        for (int d = 0; d < DIN; ++d) acc = fmaf(W[j * DIN + d], v[d], acc);
        y[(size_t)i * DOUT + j] = fmaxf(acc, 0.0f);
    }
}

// ---------------------------------------------------------------- Set2Set (one thread per pathway)
// D=3 fixed. readout[p*6 + 0..2] = h, [3..5] = r   (q_star = concat(h, r))
__global__ void set2set_k(const float* __restrict__ x2,      // [N,3]
                          const int* __restrict__ pnodes,    // [P,K]
                          const float* __restrict__ w_ih,    // [12,6]
                          const float* __restrict__ w_hh,    // [12,3]
                          const float* __restrict__ b_ih,    // [12]
                          const float* __restrict__ b_hh,    // [12]
                          float* __restrict__ readout,       // [P,6]
                          int P, int K) {
    int p = blockIdx.x * blockDim.x + threadIdx.x;
    if (p >= P) return;
    const int* nodes = pnodes + (size_t)p * K;

    float q[6] = {0, 0, 0, 0, 0, 0};
    float hs[3] = {0, 0, 0};
    float cs[3] = {0, 0, 0};

    for (int iter = 0; iter < 2; ++iter) {
        // LSTM cell, torch gate order (i,f,g,o)
        float gates[12];
#pragma unroll
        for (int j = 0; j < 12; ++j) {
            float g = b_ih[j] + b_hh[j];
#pragma unroll
            for (int t = 0; t < 6; ++t) g = fmaf(q[t], w_ih[j * 6 + t], g);
#pragma unroll
            for (int t = 0; t < 3; ++t) g = fmaf(hs[t], w_hh[j * 3 + t], g);
            gates[j] = g;
        }
#pragma unroll
        for (int d = 0; d < 3; ++d) {
            float ig = sigmoidf_(gates[d]);
            float fg = sigmoidf_(gates[3 + d]);
            float gg = tanhf(gates[6 + d]);
            float og = sigmoidf_(gates[9 + d]);
            cs[d] = fg * cs[d] + ig * gg;
            hs[d] = og * tanhf(cs[d]);
        }
        // attention: e_k = feat_k . hs ; softmax ; r = sum alpha_k feat_k
        float m = -3.402823466e+38f;
        for (int k = 0; k < K; ++k) {
            int nd = nodes[k];
            const float* f = x2 + (size_t)nd * 3;
            float e = f[0] * hs[0] + f[1] * hs[1] + f[2] * hs[2];
            m = fmaxf(m, e);
        }
        float denom = 0.0f;
        float r0 = 0.0f, r1 = 0.0f, r2 = 0.0f;
        for (int k = 0; k < K; ++k) {
            int nd = nodes[k];
            const float* f = x2 + (size_t)nd * 3;
            float e = f[0] * hs[0] + f[1] * hs[1] + f[2] * hs[2];
            float w = __expf(e - m);
            denom += w;
            r0 = fmaf(w, f[0], r0);
            r1 = fmaf(w, f[1], r1);
            r2 = fmaf(w, f[2], r2);
        }
        float inv = 1.0f / denom;
        q[0] = hs[0]; q[1] = hs[1]; q[2] = hs[2];
        q[3] = r0 * inv; q[4] = r1 * inv; q[5] = r2 * inv;
    }
#pragma unroll
    for (int j = 0; j < 6; ++j) readout[(size_t)p * 6 + j] = q[j];
}

// ---------------------------------------------------------------- lin1 GEMV with fp32 WMMA
// z[row] += sum_k W[row,k]*x[k].  Block = 16 rows, 8 waves split K.
// A (16x4 fp32, 2 VGPR): lanes 0-15 rows M, v0=K0, v1=K1; lanes 16-31 v0=K2, v1=K3.
// B replicated across all 16 columns -> result independent of B's N-layout;
// per-lane K layout mirrors A (v0=K0/K2, v1=K1/K3 by lane half).
// C column N=0: lane 0 holds rows M+0..7 in c[0..7], lane 16 holds rows M+8..15.
__global__ void lin1_wmma_k(const float* __restrict__ W,   // [Mtot, Ktot]
                            const float* __restrict__ x,   // [Ktot]
                            float* __restrict__ z,         // [Mtot] pre-zeroed
                            int Ktot) {
    const int tid  = threadIdx.x;
    // wave index is wave-uniform -> force scalar so loop control stays SALU
    const int wave = __builtin_amdgcn_readfirstlane(tid >> 5);
    const int lane = tid & 31;
    const int mbase = blockIdx.x * 16;
    const int row  = mbase + (lane & 15);
    const int hi   = lane >> 4;                 // 0: K0/K1, 1: K2/K3
    const int kPerWave = Ktot >> 3;             // 8 waves, 1500 K each
    const int chunks = kPerWave / 20;           // 75 chunks of 5 WMMAs (20 K)

    const float* wp = W + (size_t)row * Ktot + wave * kPerWave + 2 * hi;
    const float* xp = x + wave * kPerWave + 2 * hi;

    v8f c = {0.0f, 0.0f, 0.0f, 0.0f, 0.0f, 0.0f, 0.0f, 0.0f};
    for (int ch = 0; ch < chunks; ++ch) {
        // one streaming prefetch per chunk, ~2KB ahead of the weight row
        __builtin_prefetch(wp + 512, 0, 1);
        // 10 independent loads first -> clause-grouped, overlap with WMMAs
        v2f a0 = *(const v2f*)(wp +  0);
        v2f a1 = *(const v2f*)(wp +  4);
        v2f a2 = *(const v2f*)(wp +  8);
        v2f a3 = *(const v2f*)(wp + 12);
        v2f a4 = *(const v2f*)(wp + 16);
        v2f b0 = *(const v2f*)(xp +  0);
        v2f b1 = *(const v2f*)(xp +  4);
        v2f b2 = *(const v2f*)(xp +  8);
        v2f b3 = *(const v2f*)(xp + 12);
        v2f b4 = *(const v2f*)(xp + 16);
        // 8 args: (neg_a, A, neg_b, B, c_mod, C, reuse_a, reuse_b)
        c = __builtin_amdgcn_wmma_f32_16x16x4_f32(false, a0, false, b0, (short)0, c, false, false);
        c = __builtin_amdgcn_wmma_f32_16x16x4_f32(false, a1, false, b1, (short)0, c, false, false);
        c = __builtin_amdgcn_wmma_f32_16x16x4_f32(false, a2, false, b2, (short)0, c, false, false);
        c = __builtin_amdgcn_wmma_f32_16x16x4_f32(false, a3, false, b3, (short)0, c, false, false);
        c = __builtin_amdgcn_wmma_f32_16x16x4_f32(false, a4, false, b4, (short)0, c, false, false);
        wp += 20;
        xp += 20;
    }
    if ((lane & 15) == 0) {
#pragma unroll
        for (int j = 0; j < 8; ++j) atomicAdd(&z[mbase + hi * 8 + j], c[j]);
    }
}

// ---------------------------------------------------------------- final: relu(z+b1) . w2 -> sigmoid
__global__ void final_k(const float* __restrict__ z, const float* __restrict__ b1,
                        const float* __restrict__ w2, const float* __restrict__ b2,
                        float* __restrict__ out, int P) {
    __shared__ float s[256];
    int tid = threadIdx.x;
    float acc = 0.0f;
    for (int i = tid; i < P; i += 256)
        acc = fmaf(fmaxf(z[i] + b1[i], 0.0f), w2[i], acc);
    s[tid] = acc;
    __syncthreads();
    for (int off = 128; off > 0; off >>= 1) {
        if (tid < off) s[tid] += s[tid + off];
        __syncthreads();
    }
    if (tid == 0) out[0] = sigmoidf_(s[0] + b2[0]);
}

// ---------------------------------------------------------------- launch
extern "C" void kernel_launch(void* const* d_in, const int* in_sizes, int n_in,
                              void* d_out, int out_size, void* d_ws, size_t ws_size,
                              hipStream_t stream) {
    const float* h       = (const float*)d_in[0];
    const float* gin_w1  = (const float*)d_in[1];
    const float* gin_b1  = (const float*)d_in[2];
    const float* gin_w2  = (const float*)d_in[3];
    const float* gin_b2  = (const float*)d_in[4];
    const float* lstm_w_ih = (const float*)d_in[5];
    const float* lstm_w_hh = (const float*)d_in[6];
    const float* lstm_b_ih = (const float*)d_in[7];
    const float* lstm_b_hh = (const float*)d_in[8];
    const float* lin1_w  = (const float*)d_in[9];
    const float* lin1_b  = (const float*)d_in[10];
    const float* lin2_w  = (const float*)d_in[11];
    const float* lin2_b  = (const float*)d_in[12];
    const int*   edge    = (const int*)d_in[13];   // [2, E] int32
    const int*   pnodes  = (const int*)d_in[14];   // [P, K] int32

    const int N = in_sizes[0] / 3;       // 100000
    const int E = in_sizes[13] / 2;      // 3200000
    const int P = in_sizes[10];          // 2000
    const int K = in_sizes[14] / P;      // 64
    const int Ktot = P * 6;              // 12000 = P*2D

    const int* src = edge;
    const int* dst = edge + E;

    // workspace layout (floats)
    float* ws      = (float*)d_ws;
    float* agg1    = ws;                            // N*3
    float* x1      = agg1 + (size_t)N * 3;          // N*6
    float* agg2    = x1   + (size_t)N * 6;          // N*6
    float* x2      = agg2 + (size_t)N * 6;          // N*3
    float* readout = x2   + (size_t)N * 3;          // P*6
    float* z       = readout + (size_t)P * 6;       // P

    const int TB = 256;
    zero_f32<<<1024, TB, 0, stream>>>(agg1, N * 3);
    zero_f32<<<1024, TB, 0, stream>>>(agg2, N * 6);
    zero_f32<<<8,    TB, 0, stream>>>(z, P);

    // GIN conv 1
    scatter_add_k<3><<<(E + TB - 1) / TB, TB, 0, stream>>>(h, src, dst, agg1, E);
    gin_apply_k<3, 6><<<(N + TB - 1) / TB, TB, 0, stream>>>(h, agg1, gin_w1, gin_b1, x1, N);
    // GIN conv 2
    scatter_add_k<6><<<(E + TB - 1) / TB, TB, 0, stream>>>(x1, src, dst, agg2, E);
    gin_apply_k<6, 3><<<(N + TB - 1) / TB, TB, 0, stream>>>(x1, agg2, gin_w2, gin_b2, x2, N);

    // Set2Set -> readout [P*6]
    set2set_k<<<(P + TB - 1) / TB, TB, 0, stream>>>(x2, pnodes, lstm_w_ih, lstm_w_hh,
                                                    lstm_b_ih, lstm_b_hh, readout, P, K);

    // lin1 GEMV via fp32 WMMA: 125 blocks of 16 rows, 8 waves splitting K
    lin1_wmma_k<<<P / 16, 256, 0, stream>>>(lin1_w, readout, z, Ktot);

    // relu + lin2 + sigmoid
    final_k<<<1, 256, 0, stream>>>(z, lin1_b, lin2_w, lin2_b, (float*)d_out, P);
}